// LightSBplus_14259291423198
// MI455X (gfx1250) — compile-verified
//
#include <hip/hip_runtime.h>
#include <math.h>

// LightSBplus fused mixture-sample kernel for MI455X (gfx1250, wave32).
//
// Bandwidth-bound: 44 B/row * 4.19M rows ~= 184 MB -> ~8 us floor at 23.3 TB/s.
// The [N,4]x[4,5] logits GEMM is mapped onto V_WMMA_F32_16X16X4_F32 with the
// parameter matrix as A (rows = potentials) and per-row features as B columns,
// so logits for a row land *in-lane* and the Gumbel argmax needs no cross-lane
// reduction. Two tiles (32 rows) per wave-iteration; ds_bpermute_b32 moves
// tile-1 logits to the upper half-wave so the epilogue (loads of gumbel/noise,
// stores of out) runs on all 32 lanes with fully coalesced accesses.

#define KPOT 5
#define EPSF 1.0f

typedef float v2f __attribute__((ext_vector_type(2)));
typedef float v8f __attribute__((ext_vector_type(8)));

__global__ __launch_bounds__(256)
void lightsb_wmma_kernel(const float* __restrict__ x,
                         const float* __restrict__ rr,
                         const float* __restrict__ log_S,
                         const float* __restrict__ log_alpha,
                         const float* __restrict__ gumbel,
                         const float* __restrict__ noise,
                         float* __restrict__ out,
                         int rows)
{
    const int  lane = threadIdx.x & 31;
    const int  m    = lane & 15;      // column / tile-row sub-index
    const bool hi   = lane >= 16;

    // ---- uniform per-potential constants (constant-index only -> registers)
    float Sx[KPOT], Sy[KPOT], Rx[KPOT], Ry[KPOT], Qx[KPOT], Qy[KPOT], LA[KPOT];
#pragma unroll
    for (int k = 0; k < KPOT; ++k) {
        const float s0 = expf(log_S[2 * k + 0]);
        const float s1 = expf(log_S[2 * k + 1]);
        Sx[k] = s0;                    Sy[k] = s1;
        Rx[k] = rr[2 * k + 0];         Ry[k] = rr[2 * k + 1];
        Qx[k] = sqrtf(EPSF * s0);      Qy[k] = sqrtf(EPSF * s1);
        LA[k] = log_alpha[k];
    }

    // ---- A matrix (16x4 f32, rows = potentials):
    //   A[k] = [ S_k0/(2e), S_k1/(2e), r_k0/e, r_k1/e ],  rows K..15 are zero.
    // ISA layout: lanes 0-15 hold (K=0,K=1), lanes 16-31 hold (K=2,K=3).
    const float inv2e = 0.5f / EPSF;
    const float inve  = 1.0f / EPSF;
    v2f a; a.x = 0.0f; a.y = 0.0f;
    if (m < KPOT) {
        if (!hi) { a.x = expf(log_S[2 * m + 0]) * inv2e;
                   a.y = expf(log_S[2 * m + 1]) * inv2e; }
        else     { a.x = rr[2 * m + 0] * inve;
                   a.y = rr[2 * m + 1] * inve; }
    }

    // ---- C matrix: row k broadcast log_alpha[k] (rows >= KPOT unused)
    v8f c = {};
#pragma unroll
    for (int k = 0; k < KPOT; ++k) c[k] = LA[k];

    const int waveId  = blockIdx.x * (blockDim.x >> 5) + (threadIdx.x >> 5);
    const int waveCnt = gridDim.x * (blockDim.x >> 5);
    const int groups  = rows >> 5;          // 32 rows per wave-iteration

    for (int g = waveId; g < groups; g += waveCnt) {
        const int base = g << 5;
        const int n0   = base + m;          // tile-0 row for this lane's B column
        const int n1   = n0 + 16;           // tile-1 row

        const float2 xv0 = *(const float2*)(x + 2 * (size_t)n0);
        const float2 xv1 = *(const float2*)(x + 2 * (size_t)n1);

        // B (4x16): lo half supplies rows K0,K1 = x^2 ; hi half rows K2,K3 = x.
        v2f b0, b1;
        b0.x = hi ? xv0.x : xv0.x * xv0.x;
        b0.y = hi ? xv0.y : xv0.y * xv0.y;
        b1.x = hi ? xv1.x : xv1.x * xv1.x;
        b1.y = hi ? xv1.y : xv1.y * xv1.y;

        // logits[k][n] for the two 16-row tiles (full-EXEC, top level of loop)
        v8f d0 = __builtin_amdgcn_wmma_f32_16x16x4_f32(
            false, a, false, b0, (short)0, c, false, false);
        v8f d1 = __builtin_amdgcn_wmma_f32_16x16x4_f32(
            false, a, false, b1, (short)0, c, false, false);

        // lane L handles row base+L. Tile-1 logits live in lanes 0-15; hand
        // them to the hi half-wave via ds_bpermute (full-EXEC, lane m source).
        float lg[KPOT];
#pragma unroll
        for (int k = 0; k < KPOT; ++k) {
            const int t = __builtin_amdgcn_ds_bpermute(4 * m, __float_as_int(d1[k]));
            lg[k] = hi ? __int_as_float(t) : d0[k];
        }

        const int    n  = base + lane;
        const float2 xr = hi ? xv1 : xv0;   // == x[n]

        // Gumbel-max categorical sample (first-max tie rule via strict >)
        const float* gp = gumbel + (size_t)n * KPOT;
        const float l0 = lg[0] + gp[0];
        const float l1 = lg[1] + gp[1];
        const float l2 = lg[2] + gp[2];
        const float l3 = lg[3] + gp[3];
        const float l4 = lg[4] + gp[4];

        int idx = 0; float best = l0;
        if (l1 > best) { best = l1; idx = 1; }
        if (l2 > best) { best = l2; idx = 2; }
        if (l3 > best) { best = l3; idx = 3; }
        if (l4 > best) { best = l4; idx = 4; }

        // select per-potential constants (cndmask chain, no scratch)
        float sx = Sx[0], sy = Sy[0], rx = Rx[0], ry = Ry[0], qx = Qx[0], qy = Qy[0];
#pragma unroll
        for (int k = 1; k < KPOT; ++k) {
            const bool p = (idx == k);
            sx = p ? Sx[k] : sx;   sy = p ? Sy[k] : sy;
            rx = p ? Rx[k] : rx;   ry = p ? Ry[k] : ry;
            qx = p ? Qx[k] : qx;   qy = p ? Qy[k] : qy;
        }

        const float2 nv = *(const float2*)(noise + 2 * (size_t)n);
        float2 o;
        o.x = fmaf(qx, nv.x, fmaf(sx, xr.x, rx));
        o.y = fmaf(qy, nv.y, fmaf(sy, xr.y, ry));
        *(float2*)(out + 2 * (size_t)n) = o;   // full-wave 256B contiguous burst

        // stream next group's gumbel lines (largest traffic component)
        const int ng = g + waveCnt;
        if (ng < groups)
            __builtin_prefetch(gumbel + (size_t)((ng << 5) + lane) * KPOT, 0, 1);
    }
}

extern "C" void kernel_launch(void* const* d_in, const int* in_sizes, int n_in,
                              void* d_out, int out_size, void* d_ws, size_t ws_size,
                              hipStream_t stream) {
    const float* x  = (const float*)d_in[0];   // [N,2]
    const float* r  = (const float*)d_in[1];   // [K,2]
    const float* lS = (const float*)d_in[2];   // [K,2]
    const float* la = (const float*)d_in[3];   // [K]
    const float* gu = (const float*)d_in[4];   // [N,K]
    const float* no = (const float*)d_in[5];   // [N,2]
    float*       out = (float*)d_out;          // [N,2]
    (void)d_ws; (void)ws_size; (void)n_in; (void)out_size;

    const int rows   = in_sizes[0] / 2;        // N
    const int groups = rows >> 5;              // 32 rows per wave-iteration
    const int wavesPerBlock = 8;               // 256 threads, wave32
    int blocks = 2048;                         // ~16K waves; 8 groups/wave at N=4.19M
    if (blocks * wavesPerBlock > groups)
        blocks = (groups + wavesPerBlock - 1) / wavesPerBlock;
    if (blocks < 1) blocks = 1;

    hipLaunchKernelGGL(lightsb_wmma_kernel, dim3(blocks), dim3(256), 0, stream,
                       x, r, lS, la, gu, no, out, rows);
}